// SynBertForNer_41016937677038
// MI455X (gfx1250) — compile-verified
//
#include <hip/hip_runtime.h>
#include <cstdint>
#include <cstddef>

// Problem constants from the reference: B=64, L=512, D=768 (float32).
#define B_DIM 64
#define L_DIM 512
#define D_DIM 768
#define NWAVES (L_DIM / 32)   // 16 wave32s in the scan block
#define COPY_THREADS 192      // 6 wave32s; 192 lanes * 16B = 3072B = one 768-float row

// gfx1250 async global<->LDS DMA builtins (ASYNCcnt-tracked). Guarded so the
// host pass (and any toolchain lacking them) falls back to a plain VGPR copy.
#if defined(__AMDGCN__) && \
    __has_builtin(__builtin_amdgcn_global_load_async_to_lds_b128) && \
    __has_builtin(__builtin_amdgcn_global_store_async_from_lds_b128) && \
    __has_builtin(__builtin_amdgcn_s_wait_asynccnt)
#define HAVE_ASYNC_LDS 1
#else
#define HAVE_ASYNC_LDS 0
#endif

#if HAVE_ASYNC_LDS
// Builtin signature (from hipcc diagnostic): params are pointers to
// 16-byte int vectors in AS1 (global / "__device__") and AS3 (LDS / "__shared__").
typedef int v4i __attribute__((vector_size(4 * sizeof(int))));
typedef __attribute__((address_space(1))) v4i* gv4i_ptr;
typedef __attribute__((address_space(3))) v4i* lv4i_ptr;
#endif

// ---------------------------------------------------------------------------
// Kernel 1: per-batch-row compaction indices via wave32 ballot scan.
//   src_idx[b][j] = token index of the j-th valid token in row b
//   counts[b]     = number of valid tokens in row b
// One block per batch row, 512 threads (16 wave32s).
// Intra-wave: ballot mask + popcount(lanemask_lt) exclusive prefix.
// Inter-wave: 16 wave totals in LDS, ONE barrier, tiny serial prefix.
// ---------------------------------------------------------------------------
__global__ __launch_bounds__(L_DIM) void compact_scan_kernel(
    const int* __restrict__ valid_ids,  // [B, L] int32
    int* __restrict__ src_idx,          // [B, L] workspace
    int* __restrict__ counts)           // [B]   workspace
{
    const int b    = blockIdx.x;
    const int t    = threadIdx.x;
    const int lane = t & 31;            // wave32 lane
    const int w    = t >> 5;            // wave index (0..15)

    const int flag = (valid_ids[b * L_DIM + t] == 1) ? 1 : 0;

    // wave32: only the low 32 bits of the ballot are populated
    const unsigned mask = (unsigned)__ballot(flag);
    const unsigned lt   = mask & ((1u << lane) - 1u);
    const int      pre  = __popc(lt);           // exclusive prefix within wave

    __shared__ int wsum[NWAVES];
    if (lane == 0) wsum[w] = __popc(mask);      // wave total
    __syncthreads();

    int base = 0;
    #pragma unroll
    for (int i = 0; i < NWAVES; ++i) {
        if (i < w) base += wsum[i];             // exclusive prefix over waves
    }

    if (flag) {
        src_idx[b * L_DIM + base + pre] = t;
    }
    if (t == 0) {
        int tot = 0;
        #pragma unroll
        for (int i = 0; i < NWAVES; ++i) tot += wsum[i];
        counts[b] = tot;
    }
}

// ---------------------------------------------------------------------------
// Kernel 2: gather-formulated row copy (one block per output row).
//   out[b][r] = seq[b][src_idx[b][r]]  if r <  counts[b]
//   out[b][r] = 0                      if r >= counts[b]
// Copy path uses the CDNA5 async LDS DMA engine: global -> LDS -> global,
// 16 bytes per lane, ASYNCcnt-synchronized. Zero path is pure b128 stores.
// ---------------------------------------------------------------------------
__global__ __launch_bounds__(COPY_THREADS) void compact_copy_kernel(
    const float* __restrict__ seq,      // [B, L, D]
    const int* __restrict__ src_idx,    // [B, L]
    const int* __restrict__ counts,     // [B]
    float* __restrict__ out)            // [B, L, D]
{
    const int r = blockIdx.x;           // output row within batch
    const int b = blockIdx.y;           // batch index
    const int t = threadIdx.x;          // 0..191, one 16B chunk per lane

    __shared__ __align__(16) unsigned char lbuf[COPY_THREADS * 16];

    const int cnt = counts[b];
    float* dst_row = out + ((size_t)b * L_DIM + r) * D_DIM;

    if (r < cnt) {
        const int src = src_idx[b * L_DIM + r];
        const float* src_row = seq + ((size_t)b * L_DIM + src) * D_DIM;

#if HAVE_ASYNC_LDS
        // DMA 16B/lane: global -> LDS (no VGPR staging)
        __builtin_amdgcn_global_load_async_to_lds_b128(
            (gv4i_ptr)(void*)(src_row + t * 4),
            (lv4i_ptr)(lbuf + t * 16),
            0, 0);
        // LDS writes must complete before the async store reads them
        __builtin_amdgcn_s_wait_asynccnt(0);
        // DMA 16B/lane: LDS -> global
        __builtin_amdgcn_global_store_async_from_lds_b128(
            (gv4i_ptr)(void*)(dst_row + t * 4),
            (lv4i_ptr)(lbuf + t * 16),
            0, 0);
        __builtin_amdgcn_s_wait_asynccnt(0);
#else
        // Fallback: direct float4 copy through VGPRs
        const float4 v = *(reinterpret_cast<const float4*>(src_row) + t);
        *(reinterpret_cast<float4*>(dst_row) + t) = v;
        (void)lbuf;
#endif
    } else {
        // Tail rows: zero-fill (reference output is zero-initialized)
        float4 z;
        z.x = 0.0f; z.y = 0.0f; z.z = 0.0f; z.w = 0.0f;
        *(reinterpret_cast<float4*>(dst_row) + t) = z;
    }
}

// ---------------------------------------------------------------------------
// Host-side launch. Inputs (setup_inputs order):
//   d_in[0] = sequence_output  float32 [B, L, D]
//   d_in[1] = valid_ids        int32   [B, L]
// Output: float32 [B, L, D]
// Workspace layout: [0, 256)        -> counts (64 ints, padded)
//                   [256, 256+128K) -> src_idx (B*L ints)
// ---------------------------------------------------------------------------
extern "C" void kernel_launch(void* const* d_in, const int* in_sizes, int n_in,
                              void* d_out, int out_size, void* d_ws, size_t ws_size,
                              hipStream_t stream) {
    (void)in_sizes; (void)n_in; (void)out_size; (void)ws_size;

    const float* seq   = (const float*)d_in[0];
    const int*   valid = (const int*)d_in[1];
    float*       out   = (float*)d_out;

    int* counts  = (int*)d_ws;                 // 64 ints (uses first 256 B)
    int* src_idx = (int*)d_ws + 64;            // B*L ints = 128 KB

    compact_scan_kernel<<<dim3(B_DIM), dim3(L_DIM), 0, stream>>>(
        valid, src_idx, counts);

    compact_copy_kernel<<<dim3(L_DIM, B_DIM), dim3(COPY_THREADS), 0, stream>>>(
        seq, src_idx, counts, out);
}